// EmbeddingExtractor_9156870275442
// MI455X (gfx1250) — compile-verified
//
#include <hip/hip_runtime.h>
#include <cstddef>

// GraphSAGE 3-layer on gfx1250.
// N=100000 nodes, E=3.2M edges, C: 128 -> 128(relu) -> 128(relu) -> 64.
// Memory-bound on gather/scatter (~5GB @ 23.3TB/s); GEMMs done with
// v_wmma_f32_16x16x4_f32 (fp32, matching reference precision).

typedef float v2f __attribute__((ext_vector_type(2)));
typedef float v8f __attribute__((ext_vector_type(8)));

#define IN_C 128

// ---------------- utility kernels ----------------

__global__ void zero4_kernel(float4* __restrict__ p, int n4) {
    int i = blockIdx.x * blockDim.x + threadIdx.x;
    if (i < n4) p[i] = make_float4(0.f, 0.f, 0.f, 0.f);
}

__global__ void degree_kernel(const int* __restrict__ dst, float* __restrict__ cnt, int nEdges) {
    int i = blockIdx.x * blockDim.x + threadIdx.x;
    if (i < nEdges) atomicAdd(&cnt[dst[i]], 1.0f);
}

__global__ void invert_kernel(float* __restrict__ cnt, int n) {
    int i = blockIdx.x * blockDim.x + threadIdx.x;
    if (i < n) cnt[i] = 1.0f / fmaxf(cnt[i], 1.0f);
}

// ---------------- scatter-add (segment sum) ----------------
// One wave32 per edge; each lane moves one float4 chunk of the 128-float row
// (32 lanes x 16B = full 512B row, coalesced gather), then 4 f32 atomics.
__global__ void scatter_add_kernel(const float* __restrict__ feat,
                                   const int* __restrict__ src,
                                   const int* __restrict__ dst,
                                   float* __restrict__ agg,
                                   int nEdges) {
    int gwave = (blockIdx.x * blockDim.x + threadIdx.x) >> 5;
    int lane  = threadIdx.x & 31;
    if (gwave >= nEdges) return;
    int s = src[gwave];
    int d = dst[gwave];
    float4 v = ((const float4*)(feat + (size_t)s * IN_C))[lane];
    float* base = agg + (size_t)d * IN_C + lane * 4;
    atomicAdd(base + 0, v.x);
    atomicAdd(base + 1, v.y);
    atomicAdd(base + 2, v.z);
    atomicAdd(base + 3, v.w);
}

// ---------------- fused SAGE linear via WMMA f32 16x16x4 ----------------
// out[16 rows x OUT cols] = relu( mean*WlT + bias + x*WrT )
// Block: 16 node rows, (OUT/16) waves, wave w owns output cols [16w,16w+16).
// A-frag (16x4 f32): lane<16 -> M=lane, K={k,k+1}; lane>=16 -> K={k+2,k+3}.
// B-frag (4x16 f32): v0 = B[k+khalf][n], v1 = B[k+khalf+1][n], n=lane&15;
//   B[k][o] = W[o][k] (W row-major [OUT,IN]) -> contiguous float2 per lane.
template <int OUT, bool RELU>
__global__ void sage_linear_wmma_kernel(const float* __restrict__ x,
                                        const float* __restrict__ agg,
                                        const float* __restrict__ inv_deg,
                                        const float* __restrict__ Wl,
                                        const float* __restrict__ bl,
                                        const float* __restrict__ Wr,
                                        float* __restrict__ out) {
    constexpr int LDS_STRIDE = 132;  // 132 % 64 == 4 -> conflict-free ds_load_b64
    __shared__ float sMean[16 * LDS_STRIDE];
    __shared__ float sX[16 * LDS_STRIDE];

    const int row0 = blockIdx.x * 16;
    const int tid  = threadIdx.x;

    // Stage 16x128 x-tile and inv_deg-scaled 16x128 mean-tile into LDS.
    for (int i = tid; i < 16 * (IN_C / 4); i += blockDim.x) {
        int r = i / (IN_C / 4);
        int c = (i % (IN_C / 4)) * 4;
        int row = row0 + r;
        float4 vx = *(const float4*)(x + (size_t)row * IN_C + c);
        float4 va = *(const float4*)(agg + (size_t)row * IN_C + c);
        float s = inv_deg[row];
        *(float4*)(&sX[r * LDS_STRIDE + c]) = vx;
        float4 vm = make_float4(va.x * s, va.y * s, va.z * s, va.w * s);
        *(float4*)(&sMean[r * LDS_STRIDE + c]) = vm;
    }
    __syncthreads();

    const int wave = tid >> 5;
    const int lane = tid & 31;
    const int colbase = wave * 16;
    const int n = lane & 15;
    const int khalf = (lane >> 4) * 2;

    v8f acc = {};
    const float* WlRow = Wl + (size_t)(colbase + n) * IN_C;
    const float* WrRow = Wr + (size_t)(colbase + n) * IN_C;

#pragma unroll 4
    for (int k = 0; k < IN_C; k += 4) {
        int kk = k + khalf;
        v2f a_m = *(const v2f*)(&sMean[n * LDS_STRIDE + kk]);
        v2f b_l = *(const v2f*)(WlRow + kk);
        acc = __builtin_amdgcn_wmma_f32_16x16x4_f32(
            /*neg_a=*/false, a_m, /*neg_b=*/false, b_l,
            /*c_mod=*/(short)0, acc, /*reuse_a=*/false, /*reuse_b=*/false);
        v2f a_x = *(const v2f*)(&sX[n * LDS_STRIDE + kk]);
        v2f b_r = *(const v2f*)(WrRow + kk);
        acc = __builtin_amdgcn_wmma_f32_16x16x4_f32(
            false, a_x, false, b_r, (short)0, acc, false, false);
    }

    // C/D layout: lane, vgpr j -> row = j + 8*(lane>>4), col = lane&15.
    const float bias = bl[colbase + n];
    const int hi = lane >> 4;
#pragma unroll
    for (int j = 0; j < 8; ++j) {
        int row = row0 + j + 8 * hi;
        float v = acc[j] + bias;
        if (RELU) v = fmaxf(v, 0.f);
        out[(size_t)row * OUT + colbase + n] = v;
    }
}

// ---------------- host orchestration ----------------

extern "C" void kernel_launch(void* const* d_in, const int* in_sizes, int n_in,
                              void* d_out, int out_size, void* d_ws, size_t ws_size,
                              hipStream_t stream) {
    const float* x    = (const float*)d_in[0];
    const int*   ei   = (const int*)d_in[1];     // [2, E] flat
    const float* Wl1  = (const float*)d_in[2];
    const float* bl1  = (const float*)d_in[3];
    const float* Wr1  = (const float*)d_in[4];
    const float* Wl2  = (const float*)d_in[5];
    const float* bl2  = (const float*)d_in[6];
    const float* Wr2  = (const float*)d_in[7];
    const float* Wl3  = (const float*)d_in[8];
    const float* bl3  = (const float*)d_in[9];
    const float* Wr3  = (const float*)d_in[10];

    const int N = in_sizes[0] / IN_C;
    const int E = in_sizes[1] / 2;
    const int* src = ei;
    const int* dst = ei + E;

    float* inv_deg = (float*)d_ws;                     // N
    float* agg     = inv_deg + N;                      // N*128
    float* h1      = agg + (size_t)N * IN_C;           // N*128
    float* h2      = h1 + (size_t)N * IN_C;            // N*128
    float* outp    = (float*)d_out;                    // N*64

    const int feat4 = (N * IN_C) / 4;
    const int zb    = (feat4 + 255) / 256;
    const int eb    = (E + 255) / 256;
    const int nb    = (N + 255) / 256;
    const int swb   = ((E * 32) + 255) / 256;          // one wave per edge
    const int rows  = N / 16;                          // N divisible by 16

    // degree -> inverse (shared by all layers)
    zero4_kernel<<<( (N/4) + 255 ) / 256, 256, 0, stream>>>((float4*)inv_deg, N / 4);
    degree_kernel<<<eb, 256, 0, stream>>>(dst, inv_deg, E);
    invert_kernel<<<nb, 256, 0, stream>>>(inv_deg, N);

    // Layer 1: x -> h1
    zero4_kernel<<<zb, 256, 0, stream>>>((float4*)agg, feat4);
    scatter_add_kernel<<<swb, 256, 0, stream>>>(x, src, dst, agg, E);
    sage_linear_wmma_kernel<128, true><<<rows, 256, 0, stream>>>(
        x, agg, inv_deg, Wl1, bl1, Wr1, h1);

    // Layer 2: h1 -> h2
    zero4_kernel<<<zb, 256, 0, stream>>>((float4*)agg, feat4);
    scatter_add_kernel<<<swb, 256, 0, stream>>>(h1, src, dst, agg, E);
    sage_linear_wmma_kernel<128, true><<<rows, 256, 0, stream>>>(
        h1, agg, inv_deg, Wl2, bl2, Wr2, h2);

    // Layer 3: h2 -> out (N x 64, no relu)
    zero4_kernel<<<zb, 256, 0, stream>>>((float4*)agg, feat4);
    scatter_add_kernel<<<swb, 256, 0, stream>>>(h2, src, dst, agg, E);
    sage_linear_wmma_kernel<64, false><<<rows, 128, 0, stream>>>(
        h2, agg, inv_deg, Wl3, bl3, Wr3, outp);
}